// _GATLayer_3908420240274
// MI455X (gfx1250) — compile-verified
//
#include <hip/hip_runtime.h>
#include <hip/hip_bf16.h>
#include <math.h>

// ---------------- problem constants (match reference) ----------------
#define N_NODES   50000
#define N_EDGES   640000
#define IN_DIM    128
#define OUT_DIM   128
#define EDGE_DIM  16
#define N_HEADS   4
#define HEAD_DIM  32
#define LN_EPS    1e-5f

typedef __attribute__((ext_vector_type(2))) float v2f;
typedef __attribute__((ext_vector_type(8))) float v8f;

// ---------------------------------------------------------------------
// K0: init — zero out & denom, set running-max buffer to -1e9 (matches
//     jnp.maximum(segment_max, -1e9) for empty segments).
// ---------------------------------------------------------------------
__global__ __launch_bounds__(256) void k_init(float* __restrict__ out,
                                              float* __restrict__ denom,
                                              float* __restrict__ maxv) {
    int i = blockIdx.x * blockDim.x + threadIdx.x;
    if (i < N_NODES * OUT_DIM) out[i] = 0.0f;
    if (i < N_NODES * N_HEADS) { denom[i] = 0.0f; maxv[i] = -1e9f; }
}

// ---------------------------------------------------------------------
// K1: Wh = h @ W^T via V_WMMA_F32_16X16X4_F32.
// One wave per 16-row tile of h; computes all 128 output columns.
// A layout (16x4 f32): lane[3:0] = M, lane[4] selects K-pair {0,1}/{2,3},
//   VGPR0 = K even, VGPR1 = K odd.
// B layout (4x16 f32): lane[3:0] = N, lane[4] selects K-pair, mirrored.
// C/D (16x16 f32): VGPR r -> row M = r + 8*lane[4], lane[3:0] = N.
// ---------------------------------------------------------------------
__global__ __launch_bounds__(32) void k_gemm_wmma(const float* __restrict__ h,
                                                  const float* __restrict__ W,
                                                  float* __restrict__ Wh) {
    const int m0   = blockIdx.x * 16;
    const int lane = threadIdx.x;       // 0..31
    const int half = lane >> 4;         // 0/1 -> K pair select
    const int l16  = lane & 15;

    v8f acc[8] = {};                    // 8 N-tiles of 16 cols -> 128 cols

    const int mA = m0 + l16;            // A-matrix row for this lane
    const float* __restrict__ hrow = h + (size_t)mA * IN_DIM;

    for (int kc = 0; kc < IN_DIM; kc += 4) {
        const int k = kc + 2 * half;
        v2f a;
        a.x = hrow[k];
        a.y = hrow[k + 1];
#pragma unroll
        for (int nt = 0; nt < 8; ++nt) {
            const int n = nt * 16 + l16;           // output column
            v2f b;                                  // B[k][n] = W[n][k]
            b.x = W[(size_t)n * IN_DIM + k];
            b.y = W[(size_t)n * IN_DIM + k + 1];
            acc[nt] = __builtin_amdgcn_wmma_f32_16x16x4_f32(
                false, a, false, b, (short)0, acc[nt], false, false);
        }
    }

    const int mbase = m0 + (half ? 8 : 0);
#pragma unroll
    for (int nt = 0; nt < 8; ++nt) {
        const int n = nt * 16 + l16;
#pragma unroll
        for (int r = 0; r < 8; ++r) {
            Wh[(size_t)(mbase + r) * OUT_DIM + n] = acc[nt][r];
        }
    }
}

// ---------------------------------------------------------------------
// K2: per-(node,head) attention half-scores:
//   s_src[n,h] = sum_d a[h, d]      * Wh[n, h*32+d]
//   s_dst[n,h] = sum_d a[h, 32+d]   * Wh[n, h*32+d]
// ---------------------------------------------------------------------
__global__ __launch_bounds__(256) void k_node_scores(const float* __restrict__ Wh,
                                                     const float* __restrict__ a,
                                                     float* __restrict__ ssrc,
                                                     float* __restrict__ sdst) {
    int idx = blockIdx.x * blockDim.x + threadIdx.x;   // node*4 + head
    if (idx >= N_NODES * N_HEADS) return;
    const int n  = idx >> 2;
    const int hd = idx & 3;
    const float* __restrict__ w  = Wh + (size_t)n * OUT_DIM + hd * HEAD_DIM;
    const float* __restrict__ av = a + hd * (2 * HEAD_DIM);
    float s0 = 0.0f, s1 = 0.0f;
#pragma unroll
    for (int d = 0; d < HEAD_DIM; ++d) {
        const float v = w[d];
        s0 = fmaf(av[d], v, s0);
        s1 = fmaf(av[HEAD_DIM + d], v, s1);
    }
    ssrc[idx] = s0;
    sdst[idx] = s1;
}

// float atomic-max via int/uint monotonic bit trick (works for mixed signs)
__device__ __forceinline__ void atomicMaxF32(float* addr, float val) {
    if (val >= 0.0f) atomicMax((int*)addr, __float_as_int(val));
    else             atomicMin((unsigned int*)addr, __float_as_uint(val));
}

// ---------------------------------------------------------------------
// K3: per-edge logits: leaky_relu(s_src[src]+s_dst[dst]) + W_e . edge_feat
//     + running segment-max into maxv[dst].
// ---------------------------------------------------------------------
__global__ __launch_bounds__(256) void k_edge_logits(const int* __restrict__ ei,
                                                     const float* __restrict__ ef,
                                                     const float* __restrict__ We,
                                                     const float* __restrict__ ssrc,
                                                     const float* __restrict__ sdst,
                                                     float* __restrict__ attn,
                                                     float* __restrict__ maxv) {
    int e = blockIdx.x * blockDim.x + threadIdx.x;
    if (e >= N_EDGES) return;
    const int s = ei[e];
    const int d = ei[N_EDGES + e];

    const float4* __restrict__ efv = (const float4*)(ef + (size_t)e * EDGE_DIM);
    const float4 f0 = efv[0], f1 = efv[1], f2 = efv[2], f3 = efv[3];

#pragma unroll
    for (int hd = 0; hd < N_HEADS; ++hd) {
        float logit = ssrc[s * N_HEADS + hd] + sdst[d * N_HEADS + hd];
        logit = (logit > 0.0f) ? logit : 0.2f * logit;

        const float4* __restrict__ wv = (const float4*)(We + hd * EDGE_DIM);
        const float4 w0 = wv[0], w1 = wv[1], w2 = wv[2], w3 = wv[3];
        float dot = f0.x*w0.x + f0.y*w0.y + f0.z*w0.z + f0.w*w0.w
                  + f1.x*w1.x + f1.y*w1.y + f1.z*w1.z + f1.w*w1.w
                  + f2.x*w2.x + f2.y*w2.y + f2.z*w2.z + f2.w*w2.w
                  + f3.x*w3.x + f3.y*w3.y + f3.z*w3.z + f3.w*w3.w;

        const float v = logit + dot;
        attn[e * N_HEADS + hd] = v;
        atomicMaxF32(&maxv[d * N_HEADS + hd], v);
    }
}

// ---------------------------------------------------------------------
// K4: exp_v = exp(attn - maxv[dst]); denom[dst] += exp_v  (in-place attn)
// ---------------------------------------------------------------------
__global__ __launch_bounds__(256) void k_edge_exp(const int* __restrict__ ei,
                                                  float* __restrict__ attn,
                                                  const float* __restrict__ maxv,
                                                  float* __restrict__ denom) {
    int idx = blockIdx.x * blockDim.x + threadIdx.x;   // e*4 + head
    if (idx >= N_EDGES * N_HEADS) return;
    const int e  = idx >> 2;
    const int hd = idx & 3;
    const int d  = ei[N_EDGES + e];
    const float ev = __expf(attn[idx] - maxv[d * N_HEADS + hd]);
    attn[idx] = ev;
    atomicAdd(&denom[d * N_HEADS + hd], ev);
}

// ---------------------------------------------------------------------
// K5: aggregation: out[dst] += alpha * Wh[src].
// 32 threads (one wave) per edge, 4 dims per thread (float4 gather).
// Wh (25.6MB) + out (25.6MB) stay resident in the 192MB L2.
// ---------------------------------------------------------------------
__global__ __launch_bounds__(256) void k_aggregate(const int* __restrict__ ei,
                                                   const float* __restrict__ attn,
                                                   const float* __restrict__ denom,
                                                   const float* __restrict__ Wh,
                                                   float* __restrict__ out) {
    int t = blockIdx.x * blockDim.x + threadIdx.x;
    const int e = t >> 5;
    if (e >= N_EDGES) return;
    const int lane = t & 31;
    const int d0   = lane * 4;          // dims d0..d0+3
    const int hd   = lane >> 3;         // head = d0/32

    const int s = ei[e];
    const int d = ei[N_EDGES + e];

    const float alpha = attn[e * N_HEADS + hd] /
                        (denom[d * N_HEADS + hd] + 1e-9f);

    const float4 w = *(const float4*)(Wh + (size_t)s * OUT_DIM + d0);
    float* __restrict__ o = out + (size_t)d * OUT_DIM + d0;
    atomicAdd(o + 0, alpha * w.x);
    atomicAdd(o + 1, alpha * w.y);
    atomicAdd(o + 2, alpha * w.z);
    atomicAdd(o + 3, alpha * w.w);
}

// ---------------------------------------------------------------------
// K6: GELU (exact erf) + LayerNorm, in place on d_out.
// One wave32 per node; 4 values per lane; __shfl_xor wave reductions.
// ---------------------------------------------------------------------
__device__ __forceinline__ float gelu_exact(float v) {
    return 0.5f * v * (1.0f + erff(v * 0.70710678118654752f));
}

__global__ __launch_bounds__(256) void k_finalize(float* __restrict__ out,
                                                  const float* __restrict__ scale,
                                                  const float* __restrict__ bias) {
    const int gt   = blockIdx.x * blockDim.x + threadIdx.x;
    const int node = gt >> 5;
    if (node >= N_NODES) return;
    const int lane = gt & 31;
    float* __restrict__ row = out + (size_t)node * OUT_DIM;

    float4 x = *(float4*)(row + lane * 4);
    x.x = gelu_exact(x.x);
    x.y = gelu_exact(x.y);
    x.z = gelu_exact(x.z);
    x.w = gelu_exact(x.w);

    float sum = x.x + x.y + x.z + x.w;
#pragma unroll
    for (int off = 16; off > 0; off >>= 1) sum += __shfl_xor(sum, off, 32);
    const float mu = sum * (1.0f / OUT_DIM);

    const float dx = x.x - mu, dy = x.y - mu, dz = x.z - mu, dw = x.w - mu;
    float ssq = dx*dx + dy*dy + dz*dz + dw*dw;
#pragma unroll
    for (int off = 16; off > 0; off >>= 1) ssq += __shfl_xor(ssq, off, 32);
    const float inv = rsqrtf(ssq * (1.0f / OUT_DIM) + LN_EPS);

    const float4 sc = *(const float4*)(scale + lane * 4);
    const float4 bi = *(const float4*)(bias  + lane * 4);
    float4 r;
    r.x = dx * inv * sc.x + bi.x;
    r.y = dy * inv * sc.y + bi.y;
    r.z = dz * inv * sc.z + bi.z;
    r.w = dw * inv * sc.w + bi.w;
    *(float4*)(row + lane * 4) = r;
}

// ---------------------------------------------------------------------
extern "C" void kernel_launch(void* const* d_in, const int* in_sizes, int n_in,
                              void* d_out, int out_size, void* d_ws, size_t ws_size,
                              hipStream_t stream) {
    // setup_inputs order: h, edge_index, edge_feat, W, W_e, a, ln_scale, ln_bias
    const float* h    = (const float*)d_in[0];
    const int*   ei   = (const int*)  d_in[1];   // [2, E] flat
    const float* ef   = (const float*)d_in[2];
    const float* W    = (const float*)d_in[3];
    const float* We   = (const float*)d_in[4];
    const float* a    = (const float*)d_in[5];
    const float* lns  = (const float*)d_in[6];
    const float* lnb  = (const float*)d_in[7];
    float* out = (float*)d_out;

    // workspace layout (floats)
    float* ws   = (float*)d_ws;
    float* Wh   = ws;                               // N*128
    float* ssrc = Wh   + (size_t)N_NODES * OUT_DIM; // N*4
    float* sdst = ssrc + (size_t)N_NODES * N_HEADS; // N*4
    float* maxv = sdst + (size_t)N_NODES * N_HEADS; // N*4
    float* den  = maxv + (size_t)N_NODES * N_HEADS; // N*4
    float* attn = den  + (size_t)N_NODES * N_HEADS; // E*4

    const int B = 256;

    k_init<<<(N_NODES * OUT_DIM + B - 1) / B, B, 0, stream>>>(out, den, maxv);

    k_gemm_wmma<<<N_NODES / 16, 32, 0, stream>>>(h, W, Wh);

    k_node_scores<<<(N_NODES * N_HEADS + B - 1) / B, B, 0, stream>>>(Wh, a, ssrc, sdst);

    k_edge_logits<<<(N_EDGES + B - 1) / B, B, 0, stream>>>(ei, ef, We, ssrc, sdst, attn, maxv);

    k_edge_exp<<<(N_EDGES * N_HEADS + B - 1) / B, B, 0, stream>>>(ei, attn, maxv, den);

    k_aggregate<<<(N_EDGES * 32 + B - 1) / B, B, 0, stream>>>(ei, attn, den, Wh, out);

    k_finalize<<<(N_NODES * 32 + B - 1) / B, B, 0, stream>>>(out, lns, lnb);
}